// HMM_86328842650107
// MI455X (gfx1250) — compile-verified
//
#include <hip/hip_runtime.h>
#include <stdint.h>

// ---------------------------------------------------------------------------
// HMM forward recurrence on MI455X (gfx1250).
//   a_t = (a_{t-1} @ transfer) * emission[:, seq[t]]
// S=1024 states, T=8192 steps, V=64.
//
// Design (latency-bound sequential chain, ~17 GFLOP total):
//  * 16 persistent blocks x 128 threads (4 wave32s). Block b owns output
//    states j in [64b, 64b+64); wave w owns 16 of those.
//  * transfer slice (1024x64 f32 = 256KB) loaded ONCE into LDS via the
//    Tensor Data Mover (tensor_load_to_lds + s_wait_tensorcnt) and kept
//    resident for all 8192 steps -- only possible with CDNA5's 320KB WGP LDS.
//  * per step: 4KB a-vector staged global->LDS, GEMV done with
//    V_WMMA_F32_16X16X4_F32: A = transfer chunk (M=16 states, K=4),
//    B = a[4k..4k+3] broadcast across N, 256 accumulating WMMAs.
//    y[j] lands along M (N columns are duplicates).
//  * grid-wide step barrier: monotonic atomic counter in d_ws, re-zeroed
//    each call by an init kernel (graph-replay deterministic).
// ---------------------------------------------------------------------------

#define S_DIM 1024
#define T_DIM 8192
#define V_DIM 64
#define NBLK 16
#define NWAVE 4
#define BLOCK_THREADS (NWAVE * 32)
#define COLS_PER_BLK 64
#define COLS_PER_WAVE 16
#define SMEM_T_FLOATS (S_DIM * COLS_PER_BLK)           // 65536 floats = 256KB
#define SMEM_BYTES ((SMEM_T_FLOATS + S_DIM) * 4)       // + 4KB staged a-vector

typedef float v2f __attribute__((ext_vector_type(2)));
typedef float v8f __attribute__((ext_vector_type(8)));
typedef unsigned int v4u __attribute__((ext_vector_type(4)));
typedef int v8i __attribute__((ext_vector_type(8)));
typedef int v4i __attribute__((ext_vector_type(4)));

__global__ void hmm_init(const float* __restrict__ initial,
                         float* __restrict__ alpha,
                         float* __restrict__ buf0,
                         unsigned* __restrict__ cnt) {
  int s = blockIdx.x * blockDim.x + threadIdx.x;
  if (s < S_DIM) {
    float v = initial[s];
    alpha[(size_t)s * (T_DIM + 1)] = v;   // alpha[:, 0]
    buf0[s] = v;
  }
  if (s == 0) *cnt = 0u;
}

__global__ __launch_bounds__(BLOCK_THREADS)
void hmm_fwd(const int* __restrict__ seq,
             const float* __restrict__ transfer,
             const float* __restrict__ emission,
             float* __restrict__ alpha,
             float* __restrict__ buf,      // double buffer: 2 x S_DIM floats
             unsigned* __restrict__ cnt) {
  extern __shared__ float smem[];
  float* smT = smem;                  // transfer slice: smT[i*64 + (j - 64*blk)]
  float* smA = smem + SMEM_T_FLOATS;  // staged a-vector (1024 floats)

  // ---- one-time TDM bulk load of this block's 1024x64 transfer slice ----
  if (threadIdx.x == 0) {
    uint64_t ga = (uint64_t)(uintptr_t)(transfer + (size_t)blockIdx.x * COLS_PER_BLK);
    uint32_t lds = (uint32_t)(uintptr_t)smT;   // low 32 bits of generic ptr = LDS byte addr
    // D# group 0: count=1, lds_addr, global_addr[56:0], type=2 at bits 127:126
    v4u g0;
    g0[0] = 1u;
    g0[1] = lds;
    g0[2] = (uint32_t)(ga & 0xFFFFFFFFu);
    g0[3] = (uint32_t)((ga >> 32) & 0x01FFFFFFu) | (2u << 30);
    // D# group 1 (constants):
    //  data_size=2 (4B) @ bits17:16, tensor_dim0=1024 @ 79:48,
    //  tensor_dim1=1024 @ 111:80, tile_dim0=64 @ 127:112,
    //  tile_dim1=1024 @ 143:128, tensor_dim0_stride=1024 @ 207:160
    v8i g1;
    g1[0] = 0x00020000; g1[1] = 0x04000000; g1[2] = 0x04000000; g1[3] = 0x00400000;
    g1[4] = 0x00000400; g1[5] = 0x00000400; g1[6] = 0;          g1[7] = 0;
    v4i z4 = {0, 0, 0, 0};                    // groups 2/3 unused (2-D tensor)
    v8i z8 = {0, 0, 0, 0, 0, 0, 0, 0};        // extra group in 6-arg builtin
    __builtin_amdgcn_tensor_load_to_lds(g0, g1, z4, z4, z8, 0);
    __builtin_amdgcn_s_wait_tensorcnt(0);
  }
  __syncthreads();

  const int wid  = threadIdx.x >> 5;
  const int lane = threadIdx.x & 31;
  const int m    = lane & 15;        // WMMA M row (output state within wave tile)
  const int h    = lane >> 4;        // half-wave select (K pairing)
  const int n0   = blockIdx.x * COLS_PER_BLK + wid * COLS_PER_WAVE;
  const int nl   = wid * COLS_PER_WAVE + m;   // column within block slice
  const size_t Tp1 = T_DIM + 1;

  for (int t = 0; t < T_DIM; ++t) {
    const float* cur = buf + (size_t)(t & 1) * S_DIM;
    float* nxt       = buf + (size_t)((t + 1) & 1) * S_DIM;

    // stage a_{t} into LDS: 1024 floats, 128 threads -> two float4 each
    {
      const float4* src = (const float4*)cur;
      float4* dst = (float4*)smA;
      dst[threadIdx.x]                 = src[threadIdx.x];
      dst[threadIdx.x + BLOCK_THREADS] = src[threadIdx.x + BLOCK_THREADS];
    }
    __syncthreads();

    // y[n0+m] = sum_i a[i] * transfer[i][n0+m], 256 x WMMA f32 16x16x4.
    // A layout (16x4 f32): v0 = {K=0 | K=2}, v1 = {K=1 | K=3}, M = lane%16.
    // B layout (4x16 f32): v0 = row{K=0 | K=2}, v1 = row{K=1 | K=3};
    //   all N columns get the same a-value -> C columns are duplicates.
    v8f acc = {0.f, 0.f, 0.f, 0.f, 0.f, 0.f, 0.f, 0.f};
#pragma unroll 4
    for (int kb = 0; kb < S_DIM / 4; ++kb) {
      const int i0 = 4 * kb + 2 * h;
      v2f A;
      A.x = smT[i0 * COLS_PER_BLK + nl];
      A.y = smT[(i0 + 1) * COLS_PER_BLK + nl];
      v2f B = *(const v2f*)(smA + i0);   // 8B-aligned (i0 even)
      acc = __builtin_amdgcn_wmma_f32_16x16x4_f32(
          false, A, false, B, (short)0, acc, false, false);
    }

    // epilogue: lane 0 holds y[n0+r] in acc[r], lane 16 holds y[n0+8+r]
    const int obs = seq[t];
    if (m == 0) {
      const int base = n0 + 8 * h;
#pragma unroll
      for (int r = 0; r < 8; ++r) {
        float y = acc[r] * emission[(size_t)(base + r) * V_DIM + obs];
        nxt[base + r] = y;
        alpha[(size_t)(base + r) * Tp1 + (t + 1)] = y;
      }
    }

    __threadfence();     // make y stores agent-visible before arrival
    __syncthreads();     // whole block done (incl. LDS reads of smA)

    // grid-wide step barrier: monotonic counter, target = (t+1)*NBLK
    if (threadIdx.x == 0)
      __hip_atomic_fetch_add(cnt, 1u, __ATOMIC_RELEASE, __HIP_MEMORY_SCOPE_AGENT);
    const unsigned target = (unsigned)(t + 1) * NBLK;
    while (__hip_atomic_load(cnt, __ATOMIC_ACQUIRE, __HIP_MEMORY_SCOPE_AGENT) < target)
      __builtin_amdgcn_s_sleep(1);
  }
}

extern "C" void kernel_launch(void* const* d_in, const int* in_sizes, int n_in,
                              void* d_out, int out_size, void* d_ws, size_t ws_size,
                              hipStream_t stream) {
  const int*   seq      = (const int*)d_in[0];
  const float* initial  = (const float*)d_in[1];
  const float* transfer = (const float*)d_in[2];
  const float* emission = (const float*)d_in[3];
  float*    alpha = (float*)d_out;
  float*    buf   = (float*)d_ws;                                   // 2*1024 f32
  unsigned* cnt   = (unsigned*)((char*)d_ws + 2 * S_DIM * sizeof(float));

  hmm_init<<<dim3((S_DIM + 255) / 256), dim3(256), 0, stream>>>(initial, alpha, buf, cnt);
  hmm_fwd<<<dim3(NBLK), dim3(BLOCK_THREADS), SMEM_BYTES, stream>>>(
      seq, transfer, emission, alpha, buf, cnt);
}